// KeyRecorder_67697274519859
// MI455X (gfx1250) — compile-verified
//
#include <hip/hip_runtime.h>
#include <math.h>

typedef __attribute__((ext_vector_type(2))) float v2f;
typedef __attribute__((ext_vector_type(8))) float v8f;

__device__ __forceinline__ v8f wmma_f32_k4(v2f a, v2f b, v8f c) {
  // V_WMMA_F32_16X16X4_F32 : D = A(16x4) * B(4x16) + C(16x16), all fp32
  return __builtin_amdgcn_wmma_f32_16x16x4_f32(
      /*neg_a=*/false, a, /*neg_b=*/false, b,
      /*c_mod=*/(short)0, c, /*reuse_a=*/false, /*reuse_b=*/false);
}

#define B_     16
#define T_     4096
#define D_     512
#define R_     64
#define LOCAL_ 20
#define SR_    10
#define NPAST_ 408            // len(range(0, 4096-20, 10))
#define NSEL_  428            // NPAST_ + LOCAL_
#define EPS_   1e-5f

// ---------------------------------------------------------------------------
// Kernel 1: comp[f,:] = LN1(relu(obs[row(f),:] @ W1 + b1)) for the 6848
// selected rows only. One wave per 16-row tile; K=512 via 128 x 4 WMMA(f32,K=4).
// ---------------------------------------------------------------------------
__global__ __launch_bounds__(32) void k1_gemm1_ln(
    const float* __restrict__ X,  const float* __restrict__ W1,
    const float* __restrict__ b1, const float* __restrict__ lg,
    const float* __restrict__ lb, float* __restrict__ comp) {
  __shared__ float tile[16 * R_];
  __shared__ float mu_s[16];
  __shared__ float rs_s[16];

  const int lane = threadIdx.x;
  const int n0   = lane & 15;
  const int hi   = lane >> 4;     // lane half: 0 or 1
  const int kb   = hi << 1;       // K sub-offset 0 or 2 (A/B f32 layout)

  // Gather: this lane streams A-row (M = n0) of the tile.
  const int f = blockIdx.x * 16 + n0;             // flat selected row
  const int b = f / NSEL_;
  const int i = f - b * NSEL_;
  const int t = (i < NPAST_) ? i * SR_ : (T_ - LOCAL_ + (i - NPAST_));
  const float* xrow = X + ((size_t)b * T_ + t) * D_;

  v8f acc0 = {}, acc1 = {}, acc2 = {}, acc3 = {};
  for (int k = 0; k < D_; k += 4) {
    v2f a;
    a.x = xrow[k + kb];
    a.y = xrow[k + kb + 1];
    const float* wk = W1 + (size_t)(k + kb) * R_;  // W1 is (512,64) row-major
    v2f bf0, bf1, bf2, bf3;
    bf0.x = wk[     n0];  bf0.y = wk[R_      + n0];
    bf1.x = wk[16 + n0];  bf1.y = wk[R_ + 16 + n0];
    bf2.x = wk[32 + n0];  bf2.y = wk[R_ + 32 + n0];
    bf3.x = wk[48 + n0];  bf3.y = wk[R_ + 48 + n0];
    acc0 = wmma_f32_k4(a, bf0, acc0);
    acc1 = wmma_f32_k4(a, bf1, acc1);
    acc2 = wmma_f32_k4(a, bf2, acc2);
    acc3 = wmma_f32_k4(a, bf3, acc3);
  }

  // bias + relu, spill 16x64 tile to LDS (C/D layout: VGPR v -> M = v + 8*hi)
  const float bn0 = b1[n0], bn1 = b1[16 + n0], bn2 = b1[32 + n0], bn3 = b1[48 + n0];
  const int mbase = hi << 3;
#pragma unroll
  for (int v = 0; v < 8; ++v) {
    const int mr = mbase + v;
    tile[mr * R_ +      n0] = fmaxf(acc0[v] + bn0, 0.f);
    tile[mr * R_ + 16 + n0] = fmaxf(acc1[v] + bn1, 0.f);
    tile[mr * R_ + 32 + n0] = fmaxf(acc2[v] + bn2, 0.f);
    tile[mr * R_ + 48 + n0] = fmaxf(acc3[v] + bn3, 0.f);
  }
  __syncthreads();

  if (lane < 16) {
    float s = 0.f, sq = 0.f;
    for (int j = 0; j < R_; ++j) { float x = tile[lane * R_ + j]; s += x; sq += x * x; }
    float mu  = s  * (1.f / R_);
    float var = sq * (1.f / R_) - mu * mu;
    mu_s[lane] = mu;
    rs_s[lane] = rsqrtf(var + EPS_);
  }
  __syncthreads();

  for (int e = lane; e < 16 * R_; e += 32) {
    const int mr = e >> 6, n = e & (R_ - 1);
    comp[(size_t)(blockIdx.x * 16 + mr) * R_ + n] =
        (tile[e] - mu_s[mr]) * rs_s[mr] * lg[n] + lb[n];
  }
}

// ---------------------------------------------------------------------------
// Kernel 2: past max-pool over 408 sampled rows, then cummax over the 20
// present rows fused with maximum(past). g has shape (16, 20, 64).
// ---------------------------------------------------------------------------
__global__ __launch_bounds__(64) void k2_pool_cummax(
    const float* __restrict__ comp, float* __restrict__ g) {
  const int b = blockIdx.x;
  const int r = threadIdx.x;
  const float* base = comp + (size_t)b * NSEL_ * R_ + r;
  float m = -INFINITY;
  for (int i = 0; i < NPAST_; ++i) m = fmaxf(m, base[(size_t)i * R_]);
  const float* pres = base + (size_t)NPAST_ * R_;
  float* go = g + (size_t)b * LOCAL_ * R_ + r;
  float run = m;
  for (int t = 0; t < LOCAL_; ++t) {
    run = fmaxf(run, pres[(size_t)t * R_]);
    go[(size_t)t * R_] = run;
  }
}

// ---------------------------------------------------------------------------
// Kernel 3: out[f,:] = LN2(relu(g[f,:] @ W2 + b2)). 320 rows -> 20 tiles.
// One wave per tile; 32 N-tiles x 16 K-steps of WMMA(f32,K=4) each.
// ---------------------------------------------------------------------------
__global__ __launch_bounds__(32) void k3_gemm2_ln(
    const float* __restrict__ G,  const float* __restrict__ W2,
    const float* __restrict__ b2, const float* __restrict__ lg,
    const float* __restrict__ lb, float* __restrict__ out) {
  __shared__ float tile[16 * D_];   // 32 KB
  __shared__ float mu_s[16];
  __shared__ float rs_s[16];

  const int lane = threadIdx.x;
  const int n0   = lane & 15;
  const int hi   = lane >> 4;
  const int kb   = hi << 1;
  const int mbase = hi << 3;

  const float* arow = G + (size_t)(blockIdx.x * 16 + n0) * R_;

  for (int nt = 0; nt < D_ / 16; ++nt) {
    const int n = nt * 16 + n0;
    v8f acc = {};
    for (int k = 0; k < R_; k += 4) {
      v2f a;
      a.x = arow[k + kb];
      a.y = arow[k + kb + 1];
      const float* wk = W2 + (size_t)(k + kb) * D_ + n;  // W2 is (64,512) row-major
      v2f bf;
      bf.x = wk[0];
      bf.y = wk[D_];
      acc = wmma_f32_k4(a, bf, acc);
    }
    const float bias = b2[n];
#pragma unroll
    for (int v = 0; v < 8; ++v)
      tile[(mbase + v) * D_ + n] = fmaxf(acc[v] + bias, 0.f);
  }
  __syncthreads();

  if (lane < 16) {
    float s = 0.f, sq = 0.f;
    for (int j = 0; j < D_; ++j) { float x = tile[lane * D_ + j]; s += x; sq += x * x; }
    float mu  = s  * (1.f / D_);
    float var = sq * (1.f / D_) - mu * mu;
    mu_s[lane] = mu;
    rs_s[lane] = rsqrtf(var + EPS_);
  }
  __syncthreads();

  for (int e = lane; e < 16 * D_; e += 32) {
    const int mr = e >> 9, n = e & (D_ - 1);
    out[(size_t)(blockIdx.x * 16 + mr) * D_ + n] =
        (tile[e] - mu_s[mr]) * rs_s[mr] * lg[n] + lb[n];
  }
}

extern "C" void kernel_launch(void* const* d_in, const int* in_sizes, int n_in,
                              void* d_out, int out_size, void* d_ws, size_t ws_size,
                              hipStream_t stream) {
  (void)in_sizes; (void)n_in; (void)out_size; (void)ws_size;
  const float* X   = (const float*)d_in[0];  // (16,4096,512)
  const float* W1  = (const float*)d_in[1];  // (512,64)
  const float* b1  = (const float*)d_in[2];  // (64,)
  const float* lg1 = (const float*)d_in[3];
  const float* lb1 = (const float*)d_in[4];
  const float* W2  = (const float*)d_in[5];  // (64,512)
  const float* b2  = (const float*)d_in[6];  // (512,)
  const float* lg2 = (const float*)d_in[7];
  const float* lb2 = (const float*)d_in[8];
  float* out = (float*)d_out;

  float* comp_ws = (float*)d_ws;                         // 16*428*64 f32 (~1.7 MB)
  float* g_ws    = comp_ws + (size_t)B_ * NSEL_ * R_;    // 16*20*64 f32 (80 KB)

  k1_gemm1_ln<<<(B_ * NSEL_) / 16, 32, 0, stream>>>(X, W1, b1, lg1, lb1, comp_ws);
  k2_pool_cummax<<<B_, R_, 0, stream>>>(comp_ws, g_ws);
  k3_gemm2_ln<<<(B_ * LOCAL_) / 16, 32, 0, stream>>>(g_ws, W2, b2, lg2, lb2, out);
}